// M13_70480413327936
// MI455X (gfx1250) — compile-verified
//
#include <hip/hip_runtime.h>

// MI455X / gfx1250 — wave32, WMMA.
// GINE GNN: edge phase = gather + tiny GEMM + relu + scatter-atomic (memory/
// atomic bound, ~2.7GB HBM traffic -> VALU kernel, atomics resolve in 192MB L2).
// Node MLPs + head = dense f32 GEMMs via V_WMMA_F32_16X16X4_F32, fused with the
// GINE eps-combine, BatchNorm stat accumulation, and BN+LeakyReLU epilogues.
// W is staged in LDS pair-interleaved so every B fragment is ONE ds_load_b64
// (no v_mov reassembly); each wave owns MT row-tiles so B frags feed MT WMMAs.

typedef __attribute__((ext_vector_type(2))) float v2f;
typedef __attribute__((ext_vector_type(8))) float v8f;

#define LEAK 0.01f
#define BN_EPS 1e-5f
#define NNODES 100000
#define NEDGES 1600000

// ---------------------------------------------------------------- utilities

__global__ __launch_bounds__(256) void zero_f32(float* __restrict__ p, long n) {
  long i = (long)blockIdx.x * blockDim.x + threadIdx.x;
  long stride = (long)gridDim.x * blockDim.x;
  for (; i < n; i += stride) p[i] = 0.0f;
}

// cat[:, 0:64] = x
__global__ __launch_bounds__(256) void scatter_x(const float* __restrict__ x,
                                                 float* __restrict__ cat, int N) {
  long i = (long)blockIdx.x * blockDim.x + threadIdx.x;
  long n = (long)N * 64;
  long stride = (long)gridDim.x * blockDim.x;
  for (; i < n; i += stride) {
    long row = i >> 6; int c = (int)(i & 63);
    cat[row * 256 + c] = x[i];
  }
}

// training-mode BN: scale = g*rsqrt(var+eps), shift = b - mu*scale
__global__ void bn_finalize(const float* __restrict__ stats,  // [C]=sum,[C..2C)=sumsq
                            const float* __restrict__ g, const float* __restrict__ b,
                            float* __restrict__ scale, float* __restrict__ shift,
                            int C, float invN) {
  int c = threadIdx.x;
  if (c < C) {
    float mu  = stats[c] * invN;
    float var = stats[C + c] * invN - mu * mu;
    float inv = rsqrtf(var + BN_EPS);
    float s   = g[c] * inv;
    scale[c] = s;
    shift[c] = b[c] - mu * s;
  }
}

// cat[:, off:off+64] = lrelu(scale*v + shift)   (outer BN of layers 0,1)
__global__ __launch_bounds__(256) void bn_lrelu_inplace(float* __restrict__ cat, int off,
                                                        const float* __restrict__ scale,
                                                        const float* __restrict__ shift,
                                                        int N) {
  long i = (long)blockIdx.x * blockDim.x + threadIdx.x;
  long n = (long)N * 64;
  long stride = (long)gridDim.x * blockDim.x;
  for (; i < n; i += stride) {
    long row = i >> 6; int c = (int)(i & 63);
    float* p = &cat[row * 256 + off + c];
    float v = scale[c] * (*p) + shift[c];
    *p = v >= 0.0f ? v : LEAK * v;
  }
}

// ------------------------------------------------- edge message + scatter-add
// msg = relu(h[src] + ea @ We + be); agg[dst] += msg.
// One wave per edge; lane l covers features {2l, 2l+1} -> coalesced 256B row
// reads of h (cat buffer, stride 256) and coalesced f32 atomics into agg (L2).
__global__ __launch_bounds__(256) void edge_msg(
    const float* __restrict__ cat, int hoff,
    const int* __restrict__ src, const int* __restrict__ dst,
    const float* __restrict__ ea, const float* __restrict__ We,
    const float* __restrict__ be, float* __restrict__ agg, int E) {
  __shared__ float Wl[16 * 64];  // 4 KB
  int tid = threadIdx.x;
  for (int idx = tid; idx < 1024; idx += 256) Wl[idx] = We[idx];
  __syncthreads();
  int lane = tid & 31, wv = tid >> 5;
  int k2 = lane * 2;
  float bx = be[k2], by = be[k2 + 1];
  long wid = (long)blockIdx.x * 8 + wv;
  long nw  = (long)gridDim.x * 8;
  for (long e = wid; e < E; e += nw) {
    float myea = (lane < 16) ? ea[e * 16 + lane] : 0.0f;
    int s = src[e], d = dst[e];
    float ax = bx, ay = by;
#pragma unroll
    for (int j = 0; j < 16; ++j) {
      float ej = __shfl(myea, j, 32);  // broadcast ea[e][j] across the wave
      ax += ej * Wl[j * 64 + k2];
      ay += ej * Wl[j * 64 + k2 + 1];
    }
    const float* hp = cat + (size_t)s * 256 + hoff + k2;
    float mx = hp[0] + ax; mx = mx > 0.0f ? mx : 0.0f;
    float my = hp[1] + ay; my = my > 0.0f ? my : 0.0f;
    unsafeAtomicAdd(&agg[(size_t)d * 64 + k2],     mx);
    unsafeAtomicAdd(&agg[(size_t)d * 64 + k2 + 1], my);
  }
}

// --------------------------------------------------------- WMMA f32 GEMM core
// out[N x NC] = f(A)[N x K] @ W[K x NC] + bias, optional column sum/sumsq stats.
// MODE 0: f(A) = A0[row*ldA + kc + c]                       (head: cat @ Wc1)
// MODE 1: f(A) = (1+eps[layer])*A0[row*ldA+c] + agg[row*64+c]   (GINE combine)
// MODE 2: f(A) = lrelu(scl[c]*A0[row*ldA+c] + shf[c])       (BN between W1,W2)
// Block = 128 threads (4 waves); each wave owns MT 16-row tiles, NC = NT*16.
// K chunked by 64 through LDS. WMMA fragment layouts per ISA 7.12.2:
//   A(16x4 f32): lanes 0-15 row=lane, vgpr0=K0,vgpr1=K1; lanes 16-31 K2,K3.
//   B(4x16 f32): lanes 0-15 col=lane, vgpr0=K0,vgpr1=K1; lanes 16-31 K2,K3.
//   C/D: vgpr r -> row r (lanes 0-15) / row 8+r (lanes 16-31), col = lane%16.
// W is pair-interleaved in LDS: Wl[kp][2c+p] = W[2kp+p][c]  -> B frag = 1 b64.
template <int NT, int MT, int MODE>
__global__ __launch_bounds__(128) void wmma_gemm(
    const float* __restrict__ A0, const float* __restrict__ agg,
    const float* __restrict__ scl, const float* __restrict__ shf,
    const float* __restrict__ W, const float* __restrict__ bias,
    const float* __restrict__ epsPtr, int layer,
    float* __restrict__ out, int ldA, int ldOut,
    float* __restrict__ stats, int N, int K) {
  const int NC = NT * 16;
  const int ROWS = MT * 64;               // rows per block (4 waves * MT * 16)
  __shared__ float Al[ROWS][66];          // padded: banks spread across rows
  __shared__ float Wl[32][2 * NC + 2];    // pair-interleaved K-chunk of W
  int tid = threadIdx.x;
  int lane = tid & 31, wv = tid >> 5;
  int m0 = blockIdx.x * ROWS;
  float alpha = 1.0f;
  if (MODE == 1) alpha = 1.0f + epsPtr[layer];

  v8f acc[MT][NT];
  v8f zf = {};
#pragma unroll
  for (int mt = 0; mt < MT; ++mt)
#pragma unroll
    for (int t = 0; t < NT; ++t) acc[mt][t] = zf;

  const int hf = lane >> 4, r15 = lane & 15;

  for (int kc = 0; kc < K; kc += 64) {
    // stage W rows kc..kc+63, interleaving consecutive-k pairs per column
    for (int idx = tid; idx < 64 * NC; idx += 128) {
      int r = idx / NC, c = idx % NC;
      Wl[r >> 1][2 * c + (r & 1)] = W[(size_t)(kc + r) * NC + c];
    }
    // stage f(A) rows m0..m0+ROWS-1, k-chunk kc..kc+63
    for (int idx = tid; idx < ROWS * 64; idx += 128) {
      int r = idx >> 6, c = idx & 63;
      int row = m0 + r;
      float v = 0.0f;
      if (row < N) {
        if (MODE == 0) {
          v = A0[(size_t)row * ldA + kc + c];
        } else if (MODE == 1) {
          v = alpha * A0[(size_t)row * ldA + c] + agg[(size_t)row * 64 + c];
        } else {
          float t = A0[(size_t)row * ldA + c];
          t = scl[c] * t + shf[c];
          v = t >= 0.0f ? t : LEAK * t;
        }
      }
      Al[r][c] = v;
    }
    __syncthreads();

#pragma unroll
    for (int kk = 0; kk < 64; kk += 4) {
      v2f a[MT];
#pragma unroll
      for (int mt = 0; mt < MT; ++mt)
        a[mt] = *(const v2f*)&Al[wv * MT * 16 + mt * 16 + r15][kk + 2 * hf];
#pragma unroll
      for (int t = 0; t < NT; ++t) {
        v2f b = *(const v2f*)&Wl[(kk >> 1) + hf][2 * (t * 16 + r15)];
#pragma unroll
        for (int mt = 0; mt < MT; ++mt)
          acc[mt][t] = __builtin_amdgcn_wmma_f32_16x16x4_f32(
              false, a[mt], false, b, (short)0, acc[mt][t], false, false);
      }
    }
    __syncthreads();
  }

  // epilogue: + bias, store, per-column BN stats
#pragma unroll
  for (int t = 0; t < NT; ++t) {
    int col = t * 16 + r15;
    float bv = bias[col];
    float s1 = 0.0f, s2 = 0.0f;
#pragma unroll
    for (int mt = 0; mt < MT; ++mt) {
#pragma unroll
      for (int r = 0; r < 8; ++r) {
        int row = m0 + wv * MT * 16 + mt * 16 + hf * 8 + r;
        float v = acc[mt][t][r] + bv;
        if (row < N) {
          out[(size_t)row * ldOut + col] = v;
          s1 += v; s2 += v * v;
        }
      }
    }
    if (stats) {
      unsafeAtomicAdd(&stats[col], s1);
      unsafeAtomicAdd(&stats[NC + col], s2);
    }
  }
}

// --------------------------------------------- head: BN+lrelu then @Wc2 + bc2
// One wave per node; lane l handles 4 of the 128 MID features (coalesced 512B).
__global__ __launch_bounds__(256) void head_out(
    const float* __restrict__ mid, const float* __restrict__ scale,
    const float* __restrict__ shift, const float* __restrict__ Wc2,
    const float* __restrict__ bc2, float* __restrict__ out, int N) {
  int lane = threadIdx.x & 31, wv = threadIdx.x >> 5;
  long wid = (long)blockIdx.x * 8 + wv;
  long nw  = (long)gridDim.x * 8;
  int c0 = lane * 4;
  for (long n = wid; n < N; n += nw) {
    const float4* mp = (const float4*)(mid + n * 128);
    float4 m = mp[lane];
    float s = 0.0f, v;
    v = scale[c0 + 0] * m.x + shift[c0 + 0]; v = v >= 0.f ? v : LEAK * v; s += v * Wc2[c0 + 0];
    v = scale[c0 + 1] * m.y + shift[c0 + 1]; v = v >= 0.f ? v : LEAK * v; s += v * Wc2[c0 + 1];
    v = scale[c0 + 2] * m.z + shift[c0 + 2]; v = v >= 0.f ? v : LEAK * v; s += v * Wc2[c0 + 2];
    v = scale[c0 + 3] * m.w + shift[c0 + 3]; v = v >= 0.f ? v : LEAK * v; s += v * Wc2[c0 + 3];
#pragma unroll
    for (int off = 16; off >= 1; off >>= 1) s += __shfl_down(s, off, 32);
    if (lane == 0) out[n] = s + bc2[0];
  }
}

// ------------------------------------------------------------------- launcher
extern "C" void kernel_launch(void* const* d_in, const int* in_sizes, int n_in,
                              void* d_out, int out_size, void* d_ws, size_t ws_size,
                              hipStream_t stream) {
  const float* x   = (const float*)d_in[0];
  const int*   ei  = (const int*)d_in[1];    // [2, E]
  const float* ea  = (const float*)d_in[2];
  const float* eps = (const float*)d_in[3];
  const float* We  = (const float*)d_in[4];
  const float* be  = (const float*)d_in[5];
  const float* W1  = (const float*)d_in[6];
  const float* b1  = (const float*)d_in[7];
  const float* g1  = (const float*)d_in[8];
  const float* bt1 = (const float*)d_in[9];
  const float* W2  = (const float*)d_in[10];
  const float* b2  = (const float*)d_in[11];
  const float* bng = (const float*)d_in[12];
  const float* bnb = (const float*)d_in[13];
  const float* Wc1 = (const float*)d_in[14];
  const float* bc1 = (const float*)d_in[15];
  const float* gc  = (const float*)d_in[16];
  const float* btc = (const float*)d_in[17];
  const float* Wc2 = (const float*)d_in[18];
  const float* bc2 = (const float*)d_in[19];
  float* out = (float*)d_out;

  const int N = NNODES, E = NEDGES;
  // workspace layout (floats): cat[N*256] | agg[N*64] | tmp1[N*64] | stats[1024]
  // mid[N*128] aliases agg+tmp1 (only used after the GNN layers finish).
  float* ws    = (float*)d_ws;
  float* cat   = ws;
  float* agg   = ws + (size_t)N * 256;
  float* tmp1  = ws + (size_t)N * 320;
  float* mid   = agg;
  float* stats = ws + (size_t)N * 384;
  // stats slots: [0..128)=layer sum/sumsq(64+64), [128..256)=outer sum/sumsq,
  // [256..320)=scale1 [320..384)=shift1 [384..448)=scale2 [448..512)=shift2,
  // [512..768)=head sum/sumsq(128+128), [768..896)=scaleH [896..1024)=shiftH.

  const float invN = 1.0f / (float)N;
  int g128 = (N + 127) / 128;   // MT=2 GEMMs: 128 rows/block
  int g64  = (N + 63) / 64;     // MT=1 head:   64 rows/block

  scatter_x<<<4096, 256, 0, stream>>>(x, cat, N);

  for (int i = 0; i < 3; ++i) {
    zero_f32<<<2048, 256, 0, stream>>>(agg, (long)N * 64);
    zero_f32<<<1, 256, 0, stream>>>(stats, 256);
    edge_msg<<<2048, 256, 0, stream>>>(cat, i * 64, ei, ei + E, ea,
                                       We + (size_t)i * 1024, be + i * 64, agg, E);
    // tmp1 = ((1+eps)*h + agg) @ W1 + b1, accumulate BN stats
    wmma_gemm<4, 2, 1><<<g128, 128, 0, stream>>>(
        cat + i * 64, agg, nullptr, nullptr, W1 + (size_t)i * 4096, b1 + i * 64,
        eps, i, tmp1, 256, 64, stats, N, 64);
    bn_finalize<<<1, 64, 0, stream>>>(stats, g1 + i * 64, bt1 + i * 64,
                                      stats + 256, stats + 320, 64, invN);
    // cat[:, (i+1)*64 ..] = lrelu(BN(tmp1)) @ W2 + b2, outer-BN stats if i<2
    wmma_gemm<4, 2, 2><<<g128, 128, 0, stream>>>(
        tmp1, nullptr, stats + 256, stats + 320, W2 + (size_t)i * 4096, b2 + i * 64,
        nullptr, 0, cat + (i + 1) * 64, 64, 256,
        (i < 2) ? (stats + 128) : nullptr, N, 64);
    if (i < 2) {
      bn_finalize<<<1, 64, 0, stream>>>(stats + 128, bng + i * 64, bnb + i * 64,
                                        stats + 384, stats + 448, 64, invN);
      bn_lrelu_inplace<<<2048, 256, 0, stream>>>(cat, (i + 1) * 64,
                                                 stats + 384, stats + 448, N);
    }
  }

  // head: mid = cat @ Wc1 + bc1 (K=256, NC=128) with BN stats
  zero_f32<<<1, 256, 0, stream>>>(stats + 512, 256);
  wmma_gemm<8, 1, 0><<<g64, 128, 0, stream>>>(
      cat, nullptr, nullptr, nullptr, Wc1, bc1, nullptr, 0,
      mid, 256, 128, stats + 512, N, 256);
  bn_finalize<<<1, 128, 0, stream>>>(stats + 512, gc, btc,
                                     stats + 768, stats + 896, 128, invN);
  head_out<<<2048, 256, 0, stream>>>(mid, stats + 768, stats + 896, Wc2, bc2, out, N);
}